// GNNSpatialModel_45475113730093
// MI455X (gfx1250) — compile-verified
//
#include <hip/hip_runtime.h>

typedef __attribute__((ext_vector_type(2))) float v2f;
typedef __attribute__((ext_vector_type(8))) float v8f;

#define C_DIM 64

// ---------- degree / norm ----------
__global__ void init_deg_kernel(float* __restrict__ deg, int n) {
    int i = blockIdx.x * blockDim.x + threadIdx.x;
    if (i < n) deg[i] = 1.0f;  // self-loop contributes 1 to in-degree
}

__global__ void accum_deg_kernel(const int* __restrict__ dst, float* __restrict__ deg, int n_edges) {
    int e = blockIdx.x * blockDim.x + threadIdx.x;
    if (e < n_edges) atomicAdd(&deg[dst[e]], 1.0f);
}

__global__ void deg_to_dinv_kernel(float* __restrict__ deg, int n) {
    int i = blockIdx.x * blockDim.x + threadIdx.x;
    if (i < n) deg[i] = rsqrtf(deg[i]);  // deg >= 1 always
}

// ---------- dense GEMM: H[N x 64] = X[N x 64] @ W[64 x 64], fp32 WMMA ----------
// One wave computes a 16-row x 64-col output tile via 4 accumulators of 16x16,
// iterating K in steps of 4 with V_WMMA_F32_16X16X4_F32.
__global__ void gemm64_wmma_kernel(const float* __restrict__ X, const float* __restrict__ W,
                                   float* __restrict__ H, int n_tiles) {
    int wave = blockIdx.x * (blockDim.x >> 5) + (threadIdx.x >> 5);
    if (wave >= n_tiles) return;            // uniform per wave -> EXEC stays all-ones
    int lane = threadIdx.x & 31;
    int m  = lane & 15;                      // A: M index, B/C/D: N index / column-in-halfwave
    int hi = lane >> 4;                      // lane half selects K pair / C row offset

    const float* xrow = X + (size_t)(wave * 16 + m) * C_DIM;

    v8f acc0 = {}, acc1 = {}, acc2 = {}, acc3 = {};

#pragma unroll
    for (int kb = 0; kb < 64; kb += 4) {
        int k0 = kb + 2 * hi;               // lanes 0-15: K={kb,kb+1}; lanes 16-31: K={kb+2,kb+3}
        v2f a;
        a.x = xrow[k0];
        a.y = xrow[k0 + 1];

        const float* w0 = W + (size_t)k0 * C_DIM + m;   // row K=k0
        const float* w1 = w0 + C_DIM;                   // row K=k0+1
        v2f b0; b0.x = w0[0];  b0.y = w1[0];
        v2f b1; b1.x = w0[16]; b1.y = w1[16];
        v2f b2; b2.x = w0[32]; b2.y = w1[32];
        v2f b3; b3.x = w0[48]; b3.y = w1[48];

        acc0 = __builtin_amdgcn_wmma_f32_16x16x4_f32(false, a, false, b0, (short)0, acc0, false, false);
        acc1 = __builtin_amdgcn_wmma_f32_16x16x4_f32(false, a, false, b1, (short)0, acc1, false, false);
        acc2 = __builtin_amdgcn_wmma_f32_16x16x4_f32(false, a, false, b2, (short)0, acc2, false, false);
        acc3 = __builtin_amdgcn_wmma_f32_16x16x4_f32(false, a, false, b3, (short)0, acc3, false, false);
    }

    // C/D layout: VGPR r, lanes 0-15 -> M=r, lanes 16-31 -> M=r+8; N = lane&15
    float* hbase = H + (size_t)(wave * 16) * C_DIM + m;
#pragma unroll
    for (int r = 0; r < 8; ++r) {
        float* orow = hbase + (size_t)(r + 8 * hi) * C_DIM;
        orow[0]  = acc0[r];
        orow[16] = acc1[r];
        orow[32] = acc2[r];
        orow[48] = acc3[r];
    }
}

// ---------- zero accumulator ----------
__global__ void zero4_kernel(float4* __restrict__ p, long long n4) {
    long long i = (long long)blockIdx.x * blockDim.x + threadIdx.x;
    if (i < n4) p[i] = make_float4(0.f, 0.f, 0.f, 0.f);
}

// ---------- edge scatter: ACC[dst] += dinv[src]*dinv[dst] * H[src]  (warp per edge) ----------
__global__ void scatter_kernel(const int* __restrict__ src, const int* __restrict__ dst,
                               const float* __restrict__ dinv, const float* __restrict__ H,
                               float* __restrict__ ACC, int n_edges) {
    int wave = blockIdx.x * (blockDim.x >> 5) + (threadIdx.x >> 5);
    if (wave >= n_edges) return;
    int lane = threadIdx.x & 31;

    int s = src[wave];
    int d = dst[wave];
    float norm = dinv[s] * dinv[d];

    float2 v = *((const float2*)(H + (size_t)s * C_DIM) + lane);  // 2 channels per lane
    float* od = ACC + (size_t)d * C_DIM + lane * 2;
    atomicAdd(od,     norm * v.x);
    atomicAdd(od + 1, norm * v.y);
}

// ---------- finalize: out = relu(ACC + dinv^2 * H + bias) ----------
__global__ void finalize_kernel(const float* __restrict__ ACC, const float* __restrict__ H,
                                const float* __restrict__ dinv, const float* __restrict__ bias,
                                float* __restrict__ out, int n_nodes) {
    int t = blockIdx.x * blockDim.x + threadIdx.x;       // one float4 per thread
    int total = n_nodes * (C_DIM / 4);
    if (t >= total) return;
    int node = t >> 4;
    int c = (t & 15) * 4;
    float di = dinv[node];
    float sl = di * di;                                   // self-loop norm
    size_t off = (size_t)node * C_DIM + c;
    float4 a = *(const float4*)(ACC + off);
    float4 h = *(const float4*)(H + off);
    float4 b = *(const float4*)(bias + c);
    float4 o;
    o.x = fmaxf(a.x + sl * h.x + b.x, 0.0f);
    o.y = fmaxf(a.y + sl * h.y + b.y, 0.0f);
    o.z = fmaxf(a.z + sl * h.z + b.z, 0.0f);
    o.w = fmaxf(a.w + sl * h.w + b.w, 0.0f);
    *(float4*)(out + off) = o;
}

extern "C" void kernel_launch(void* const* d_in, const int* in_sizes, int n_in,
                              void* d_out, int out_size, void* d_ws, size_t ws_size,
                              hipStream_t stream) {
    const float* x  = (const float*)d_in[0];
    const int*   ei = (const int*)d_in[1];
    const float* W1 = (const float*)d_in[2];
    const float* b1 = (const float*)d_in[3];
    const float* W2 = (const float*)d_in[4];
    const float* b2 = (const float*)d_in[5];
    float* out = (float*)d_out;

    const int n_nodes = in_sizes[0] / C_DIM;   // 100000
    const int n_edges = in_sizes[1] / 2;       // 1600000
    const int* src = ei;
    const int* dst = ei + n_edges;

    // workspace layout: dinv | H | ACC   (~52 MB)
    char* ws = (char*)d_ws;
    float* dinv = (float*)ws;
    size_t off = ((size_t)n_nodes * sizeof(float) + 255) & ~(size_t)255;
    float* H = (float*)(ws + off);
    off += (size_t)n_nodes * C_DIM * sizeof(float);
    float* ACC = (float*)(ws + off);

    const int n_tiles = n_nodes / 16;                 // 6250 (exact)
    const int gemm_blocks = (n_tiles + 7) / 8;        // 8 waves / 256-thread block
    const long long n_feat4 = (long long)n_nodes * (C_DIM / 4);
    const int node_blocks = (n_nodes + 255) / 256;
    const int edge_blocks_t = (n_edges + 255) / 256;  // thread-per-edge
    const int edge_blocks_w = (n_edges + 7) / 8;      // wave-per-edge
    const int feat4_blocks = (int)((n_feat4 + 255) / 256);

    // shared normalization (both layers use the same src/dst with self-loops)
    init_deg_kernel<<<node_blocks, 256, 0, stream>>>(dinv, n_nodes);
    accum_deg_kernel<<<edge_blocks_t, 256, 0, stream>>>(dst, dinv, n_edges);
    deg_to_dinv_kernel<<<node_blocks, 256, 0, stream>>>(dinv, n_nodes);

    // ---- layer 1 ----
    gemm64_wmma_kernel<<<gemm_blocks, 256, 0, stream>>>(x, W1, H, n_tiles);
    zero4_kernel<<<feat4_blocks, 256, 0, stream>>>((float4*)ACC, n_feat4);
    scatter_kernel<<<edge_blocks_w, 256, 0, stream>>>(src, dst, dinv, H, ACC, n_edges);
    finalize_kernel<<<feat4_blocks, 256, 0, stream>>>(ACC, H, dinv, b1, out, n_nodes);  // out = h1

    // ---- layer 2 (reads h1 from d_out, overwrites d_out at the end) ----
    gemm64_wmma_kernel<<<gemm_blocks, 256, 0, stream>>>(out, W2, H, n_tiles);
    zero4_kernel<<<feat4_blocks, 256, 0, stream>>>((float4*)ACC, n_feat4);
    scatter_kernel<<<edge_blocks_w, 256, 0, stream>>>(src, dst, dinv, H, ACC, n_edges);
    finalize_kernel<<<feat4_blocks, 256, 0, stream>>>(ACC, H, dinv, b2, out, n_nodes);
}